// GATv2_25924422599408
// MI455X (gfx1250) — compile-verified
//
#include <hip/hip_runtime.h>
#include <hip/hip_bf16.h>
#include <math.h>

#define N_NODES 50000
#define E_EDGES 800000
#define IN_DIM  128
#define HID     64
#define HEADS   4
#define OUT_DIM 2
#define NEG_SLOPE 0.2f

typedef float v2f __attribute__((ext_vector_type(2)));
typedef float v8f __attribute__((ext_vector_type(8)));

// ---------- helpers ----------

__device__ __forceinline__ void atomicMaxF32(float* addr, float val) {
    // monotone float-max via integer atomics (works with mixed signs, init -inf)
    if (val >= 0.0f) {
        atomicMax((int*)addr, __float_as_int(val));
    } else {
        atomicMin((unsigned int*)addr, (unsigned int)__float_as_int(val));
    }
}

__global__ void fill_f32(float* __restrict__ p, float v, long n) {
    long i = (long)blockIdx.x * blockDim.x + threadIdx.x;
    if (i < n) p[i] = v;
}

// ---------- dense dual GEMM via V_WMMA_F32_16X16X4_F32 ----------
// Cl = A @ Bl ; Cr = A @ Br  (row-major). A fragment is loaded once and fed to
// two WMMAs per K-step (two independent accumulator chains).
// K, Ncol compile-time so every global load uses an immediate offset.
// One wave owns one 16x16 tile of Cl+Cr. Block = 4 waves on 4 distinct N tiles.
template<int K, int Ncol>
__global__ void __launch_bounds__(128)
wmma_gemm_dual(const float* __restrict__ A,
               const float* __restrict__ Bl, const float* __restrict__ Br,
               float* __restrict__ Cl, float* __restrict__ Cr) {
    const int wave  = threadIdx.x >> 5;
    const int lane  = threadIdx.x & 31;
    const int hf    = lane >> 4;   // K half selector / D row-half selector
    const int mn    = lane & 15;
    const int tileM = blockIdx.x * 16;
    const int tileN = (blockIdx.y * 4 + wave) * 16;
    if (tileN >= Ncol) return;     // uniform per wave -> EXEC all-1s inside wave

    // A fragment: lane holds A[tileM+mn][k + 2*hf + {0,1}]
    const float* ap  = A  + (long)(tileM + mn) * K + 2 * hf;
    // B fragments: lane holds B[k + 2*hf + {0,1}][tileN+mn]
    const float* blp = Bl + (long)(2 * hf) * Ncol + (tileN + mn);
    const float* brp = Br + (long)(2 * hf) * Ncol + (tileN + mn);

    v8f accl = {}, accr = {};
#pragma unroll
    for (int k = 0; k < K; k += 4) {
        v2f a;  a.x  = ap[k];               a.y  = ap[k + 1];
        v2f bl; bl.x = blp[(long)k * Ncol]; bl.y = blp[(long)(k + 1) * Ncol];
        v2f br; br.x = brp[(long)k * Ncol]; br.y = brp[(long)(k + 1) * Ncol];
        accl = __builtin_amdgcn_wmma_f32_16x16x4_f32(
                   false, a, false, bl, (short)0, accl, false, false);
        accr = __builtin_amdgcn_wmma_f32_16x16x4_f32(
                   false, a, false, br, (short)0, accr, false, false);
    }
    // D layout: VGPR v -> row M = v + 8*hf, col N = tileN + mn
#pragma unroll
    for (int v = 0; v < 8; ++v) {
        const long ci = (long)(tileM + v + 8 * hf) * Ncol + (tileN + mn);
        Cl[ci] = accl[v];
        Cr[ci] = accr[v];
    }
}

// ---------- edge kernels (templated over head count) ----------

// score[e,h] = sum_d att[h,d] * leaky_relu(xl[src,h,d] + xr[dst,h,d]);  m[n,h] = max over incoming
template<int H>
__global__ void edge_score(const float* __restrict__ xl, const float* __restrict__ xr,
                           const float* __restrict__ att,
                           const long long* __restrict__ src, const long long* __restrict__ dst,
                           float* __restrict__ score, float* __restrict__ mbuf) {
    const long wid  = ((long)blockIdx.x * blockDim.x + threadIdx.x) >> 5;
    const int  lane = threadIdx.x & 31;
    if (wid >= (long)E_EDGES * H) return;
    const long e = wid / H;
    const int  h = (int)(wid - e * H);
    const long s = (long)src[e], d = (long)dst[e];
    const float* pl = xl + s * (H * HID) + h * HID;
    const float* pr = xr + d * (H * HID) + h * HID;
    const float* pa = att + h * HID;
    float acc = 0.0f;
#pragma unroll
    for (int c = 0; c < HID / 32; ++c) {
        const int i = c * 32 + lane;
        float v = pl[i] + pr[i];
        v = v > 0.0f ? v : v * NEG_SLOPE;
        acc += v * pa[i];
    }
#pragma unroll
    for (int off = 16; off > 0; off >>= 1) acc += __shfl_xor(acc, off, 32);
    if (lane == 0) {
        score[e * H + h] = acc;
        atomicMaxF32(&mbuf[d * H + h], acc);
    }
}

// w = exp(score - m[dst]); score <- w; denom[dst] += w
template<int H>
__global__ void edge_exp(const long long* __restrict__ dst, float* __restrict__ score,
                         const float* __restrict__ mbuf, float* __restrict__ denom) {
    const long i = (long)blockIdx.x * blockDim.x + threadIdx.x;
    if (i >= (long)E_EDGES * H) return;
    const long e = i / H;
    const int  h = (int)(i - e * H);
    const long d = (long)dst[e];
    const float w = expf(score[i] - mbuf[d * H + h]);
    score[i] = w;
    atomicAdd(&denom[d * H + h], w);
}

// out[dst, h, d] += (w[e,h]/denom[dst,h]) * xl[src, h, d]   (one wave per edge)
template<int H>
__global__ void edge_aggregate(const float* __restrict__ xl,
                               const long long* __restrict__ src, const long long* __restrict__ dst,
                               const float* __restrict__ w, const float* __restrict__ denom,
                               float* __restrict__ out) {
    const long e    = ((long)blockIdx.x * blockDim.x + threadIdx.x) >> 5;
    const int  lane = threadIdx.x & 31;
    if (e >= E_EDGES) return;
    const long s = (long)src[e], d = (long)dst[e];
#pragma unroll
    for (int c = 0; c < (H * HID) / 32; ++c) {
        const int v = c * 32 + lane;
        const int h = v >> 6;                          // uniform within each 32-lane chunk
        const float alpha = w[e * H + h] / denom[d * H + h];
        atomicAdd(&out[d * (long)(H * HID) + v], alpha * xl[s * (long)(H * HID) + v]);
    }
}

// acc <- elu(acc + bias[col])   (in place)
__global__ void elu_bias(float* __restrict__ acc, const float* __restrict__ bias,
                         int cols, long total) {
    const long i = (long)blockIdx.x * blockDim.x + threadIdx.x;
    if (i >= total) return;
    const int c = (int)(i % cols);
    float v = acc[i] + bias[c];
    acc[i] = v > 0.0f ? v : (expf(v) - 1.0f);
}

// out[n, 0..1] = elu(acc2[n,:] + b2) @ Wlin + blin   (one wave per node)
__global__ void final_linear(const float* __restrict__ acc2, const float* __restrict__ b2,
                             const float* __restrict__ Wlin, const float* __restrict__ blin,
                             float* __restrict__ out) {
    const long n    = ((long)blockIdx.x * blockDim.x + threadIdx.x) >> 5;
    const int  lane = threadIdx.x & 31;
    if (n >= N_NODES) return;
    float p0 = 0.0f, p1 = 0.0f;
#pragma unroll
    for (int c = 0; c < HID / 32; ++c) {
        const int d = c * 32 + lane;
        float v = acc2[n * (long)HID + d] + b2[d];
        v = v > 0.0f ? v : (expf(v) - 1.0f);
        p0 += v * Wlin[d * OUT_DIM + 0];
        p1 += v * Wlin[d * OUT_DIM + 1];
    }
#pragma unroll
    for (int off = 16; off > 0; off >>= 1) {
        p0 += __shfl_xor(p0, off, 32);
        p1 += __shfl_xor(p1, off, 32);
    }
    if (lane == 0) {
        out[n * OUT_DIM + 0] = p0 + blin[0];
        out[n * OUT_DIM + 1] = p1 + blin[1];
    }
}

// ---------- driver ----------

extern "C" void kernel_launch(void* const* d_in, const int* in_sizes, int n_in,
                              void* d_out, int out_size, void* d_ws, size_t ws_size,
                              hipStream_t stream) {
    const float*     x    = (const float*)d_in[0];
    const long long* ei   = (const long long*)d_in[1];   // int64 [2, E]
    const float*     W1l  = (const float*)d_in[2];
    const float*     W1r  = (const float*)d_in[3];
    const float*     att1 = (const float*)d_in[4];
    const float*     b1   = (const float*)d_in[5];
    const float*     W2l  = (const float*)d_in[6];
    const float*     W2r  = (const float*)d_in[7];
    const float*     att2 = (const float*)d_in[8];
    const float*     b2   = (const float*)d_in[9];
    const float*     Wlin = (const float*)d_in[10];
    const float*     blin = (const float*)d_in[11];
    const long long* src  = ei;
    const long long* dst  = ei + E_EDGES;
    float* outp = (float*)d_out;

    const int HD1 = HEADS * HID;   // 256
    // workspace layout (floats)
    float* ws  = (float*)d_ws;
    float* xl1 = ws;                          // N*256
    float* xr1 = xl1 + (long)N_NODES * HD1;   // N*256
    float* h1  = xr1 + (long)N_NODES * HD1;   // N*256  (accumulator -> hidden after elu)
    float* sc1 = h1  + (long)N_NODES * HD1;   // E*4
    float* m1  = sc1 + (long)E_EDGES * HEADS; // N*4
    float* dn1 = m1  + (long)N_NODES * HEADS; // N*4
    // layer-2 reuse of the xl1/xr1 regions
    float* xl2  = xl1;                        // N*64
    float* xr2  = xl1 + (long)N_NODES * HID;  // N*64
    float* acc2 = xl1 + (long)N_NODES * 2 * HID; // N*64
    float* sc2 = sc1;                         // E
    float* m2  = m1;                          // N
    float* dn2 = dn1;                         // N

    const int TB = 256;
    auto blocks = [](long n, int tb) { return (unsigned)((n + tb - 1) / tb); };

    // ---------------- layer 1 ----------------
    fill_f32<<<blocks((long)N_NODES * HEADS, TB), TB, 0, stream>>>(m1, -INFINITY, (long)N_NODES * HEADS);
    fill_f32<<<blocks((long)N_NODES * HEADS, TB), TB, 0, stream>>>(dn1, 0.0f, (long)N_NODES * HEADS);
    fill_f32<<<blocks((long)N_NODES * HD1,  TB), TB, 0, stream>>>(h1, 0.0f, (long)N_NODES * HD1);

    // xl1 = x @ W1l ; xr1 = x @ W1r   (M=50000, K=128, Ncol=256) — fused dual GEMM
    {
        dim3 grid(N_NODES / 16, (HD1 / 16) / 4);   // 3125 x 4, 4 waves/block = 4 N-tiles
        wmma_gemm_dual<IN_DIM, HEADS * HID><<<grid, 128, 0, stream>>>(x, W1l, W1r, xl1, xr1);
    }

    edge_score<HEADS><<<blocks((long)E_EDGES * HEADS * 32, TB), TB, 0, stream>>>(
        xl1, xr1, att1, src, dst, sc1, m1);
    edge_exp<HEADS><<<blocks((long)E_EDGES * HEADS, TB), TB, 0, stream>>>(dst, sc1, m1, dn1);
    edge_aggregate<HEADS><<<blocks((long)E_EDGES * 32, TB), TB, 0, stream>>>(
        xl1, src, dst, sc1, dn1, h1);
    elu_bias<<<blocks((long)N_NODES * HD1, TB), TB, 0, stream>>>(h1, b1, HD1, (long)N_NODES * HD1);

    // ---------------- layer 2 ----------------
    fill_f32<<<blocks((long)N_NODES, TB), TB, 0, stream>>>(m2, -INFINITY, (long)N_NODES);
    fill_f32<<<blocks((long)N_NODES, TB), TB, 0, stream>>>(dn2, 0.0f, (long)N_NODES);
    fill_f32<<<blocks((long)N_NODES * HID, TB), TB, 0, stream>>>(acc2, 0.0f, (long)N_NODES * HID);

    // xl2 = h1 @ W2l ; xr2 = h1 @ W2r   (M=50000, K=256, Ncol=64) — fused dual GEMM
    {
        dim3 grid(N_NODES / 16, 1);                // Ncol=64 -> exactly 4 N-tiles = 4 waves
        wmma_gemm_dual<HEADS * HID, HID><<<grid, 128, 0, stream>>>(h1, W2l, W2r, xl2, xr2);
    }

    edge_score<1><<<blocks((long)E_EDGES * 32, TB), TB, 0, stream>>>(
        xl2, xr2, att2, src, dst, sc2, m2);
    edge_exp<1><<<blocks((long)E_EDGES, TB), TB, 0, stream>>>(dst, sc2, m2, dn2);
    edge_aggregate<1><<<blocks((long)E_EDGES * 32, TB), TB, 0, stream>>>(
        xl2, src, dst, sc2, dn2, acc2);

    // elu + final linear fused
    final_linear<<<blocks((long)N_NODES * 32, TB), TB, 0, stream>>>(acc2, b2, Wlin, blin, outp);
}